// DynamicDirectionalConv_83339545412067
// MI455X (gfx1250) — compile-verified
//
#include <hip/hip_runtime.h>
#include <hip/hip_bf16.h>

typedef float v2f __attribute__((ext_vector_type(2)));
typedef float v8f __attribute__((ext_vector_type(8)));

#define BB    8
#define CCH   128
#define HH    128
#define WW    128
#define KS    7
#define PADR  3
#define TILEW 16
#define TILEH 32                 // 2 output rows per thread, 16 row-groups
#define HALOW (TILEW + KS - 1)   // 22
#define HALOH (TILEH + KS - 1)   // 38
#define LSTR  (HALOW + 2)        // 24, dodge bank conflicts
#define NCH   8                  // channels staged in LDS per batch
#define CGRP  32                 // channels handled per block (4 batches)

__device__ __forceinline__ int refl(int i, int n) {
    if (i < 0) i = -i;                 // jnp.pad 'reflect': -1 -> 1
    if (i >= n) i = 2 * n - 2 - i;     //  n -> n-2
    return i;
}

// ---------------------------------------------------------------------------
// Kernel 1: per-pixel gating MLP.
//   h^T(8x16) = w1_pad(16x4) x feat_pad(4x16)  via v_wmma_f32_16x16x4_f32.
//   Neurons on M, pixels on N  =>  lane p holds h[p][0..7] in acc[0..7],
//   so layer2 + softmax are per-lane VALU (no cross-lane extraction).
// ---------------------------------------------------------------------------
__global__ void gate_weights_kernel(const float* __restrict__ angle_map,
                                    const float* __restrict__ w1,
                                    const float* __restrict__ b1,
                                    const float* __restrict__ w2,
                                    const float* __restrict__ b2,
                                    float* __restrict__ wgt_out) {
    const int lane = threadIdx.x & 31;
    const int wave = threadIdx.x >> 5;
    const int waveGlobal = blockIdx.x * (blockDim.x >> 5) + wave;
    const int sub = lane & 15;
    const int pix = waveGlobal * 16 + sub;          // < B*H*W (exact multiple)

    const float th = angle_map[pix];
    const float sn = sinf(2.0f * th);
    const float cs = cosf(2.0f * th);

    const bool lo = (lane < 16);

    // A = w1 zero-padded to 16x4. f32 A layout: lanes0-15 hold K=0,1;
    // lanes16-31 hold K=2,3 (zero padding).
    v2f a;
    a.x = (lo && sub < 8) ? w1[sub * 2 + 0] : 0.0f;
    a.y = (lo && sub < 8) ? w1[sub * 2 + 1] : 0.0f;

    // B = feat padded to 4x16. VGPR0: lanes0-15 row K=0 (sin), lanes16-31 row
    // K=2 (pad=0); VGPR1: K=1 (cos) / K=3 (pad=0).
    v2f bm;
    bm.x = lo ? sn : 0.0f;
    bm.y = lo ? cs : 0.0f;

    v8f acc = {};
    acc = __builtin_amdgcn_wmma_f32_16x16x4_f32(false, a, false, bm,
                                                (short)0, acc, false, false);

    if (lo) {
        float h[8];
        #pragma unroll
        for (int n = 0; n < 8; ++n) {
            float v = acc[n] + b1[n];
            h[n] = v > 0.0f ? v : 0.0f;
        }
        float lg[4];
        #pragma unroll
        for (int d = 0; d < 4; ++d) {
            float v = b2[d];
            #pragma unroll
            for (int n = 0; n < 8; ++n) v = fmaf(w2[d * 8 + n], h[n], v);
            lg[d] = v;
        }
        const float mx = fmaxf(fmaxf(lg[0], lg[1]), fmaxf(lg[2], lg[3]));
        float e[4];
        float s = 0.0f;
        #pragma unroll
        for (int d = 0; d < 4; ++d) { e[d] = expf(lg[d] - mx); s += e[d]; }
        const float inv = 1.0f / s;
        float4 o;
        o.x = e[0] * inv; o.y = e[1] * inv; o.z = e[2] * inv; o.w = e[3] * inv;
        *(float4*)(wgt_out + 4 * (size_t)pix) = o;
    }
}

// ---------------------------------------------------------------------------
// Kernel 2: folded directional conv, 2 output rows per thread.
//   out[b,c,h,w] = sum_t K_eff[b,h,w,t] * xp[b,c, h+di(t), w+dj(t)]
//   Two vertically adjacent pixels share 6 of 8 input rows:
//   56 LDS reads per 98 FMAs -> balances the WGP's 256 B/clk LDS port
//   against 128 fp32 FMA/clk. K_eff for both pixels lives in registers.
//   Channels split over 4 block groups for occupancy.
// ---------------------------------------------------------------------------
__global__ void dirconv_kernel(const float* __restrict__ x,
                               const float* __restrict__ bker,
                               const float* __restrict__ wgt,
                               float* __restrict__ out) {
    __shared__ float s_ker[4 * KS * KS];
    __shared__ float s_x[NCH][HALOH * LSTR];   // 8 * 38*24 * 4B = 29184 B

    const int tid = threadIdx.x;
    const int wb  = blockIdx.x & 7;            // W tile   (8)
    const int hb  = (blockIdx.x >> 3) & 3;     // H tile   (4)
    const int b   = (blockIdx.x >> 5) & 7;     // batch    (8)
    const int cg  = blockIdx.x >> 8;           // channel group (4)
    const int h0 = hb * TILEH, w0 = wb * TILEW;
    const int tx  = tid & 15;
    const int tyg = tid >> 4;                  // 0..15 row groups
    const int o0 = tyg * 2;                    // two adjacent output rows
    const int hA = h0 + o0, hB = hA + 1;
    const int wcol = w0 + tx;

    if (tid < 4 * KS * KS) s_ker[tid] = bker[tid];
    const float4 gA = *(const float4*)(wgt + 4 * (((size_t)b * HH + hA) * WW + wcol));
    const float4 gB = *(const float4*)(wgt + 4 * (((size_t)b * HH + hB) * WW + wcol));
    __syncthreads();

    float k0[KS * KS], k1[KS * KS];
    #pragma unroll
    for (int t = 0; t < KS * KS; ++t) {
        const float c0 = s_ker[t], c1 = s_ker[49 + t],
                    c2 = s_ker[98 + t], c3 = s_ker[147 + t];
        k0[t] = gA.x * c0 + gA.y * c1 + gA.z * c2 + gA.w * c3;
        k1[t] = gB.x * c0 + gB.y * c1 + gB.z * c2 + gB.w * c3;
    }

    for (int cb = 0; cb < CGRP / NCH; ++cb) {
        // cooperative staged load with reflect padding
        for (int e = tid; e < NCH * HALOH * HALOW; e += 256) {
            const int cc = e / (HALOH * HALOW);
            const int r  = (e % (HALOH * HALOW)) / HALOW;
            const int cl = e % HALOW;
            const int gh = refl(h0 - PADR + r, HH);
            const int gw = refl(w0 - PADR + cl, WW);
            const int c  = cg * CGRP + cb * NCH + cc;
            s_x[cc][r * LSTR + cl] =
                x[(((size_t)b * CCH + c) * HH + gh) * WW + gw];
        }
        __syncthreads();

        // hint-prefetch next channel batch while computing this one
        if (cb + 1 < CGRP / NCH) {
            const int cc = tid >> 5;          // 0..7
            const int r  = tid & 31;
            if (r < HALOH / 2) {              // coarse row coverage (128B lines)
                const int c  = cg * CGRP + (cb + 1) * NCH + cc;
                const int gh = refl(h0 - PADR + 2 * r, HH);
                const int gw = refl(w0 - PADR, WW);
                __builtin_prefetch(&x[(((size_t)b * CCH + c) * HH + gh) * WW + gw], 0, 1);
            }
        }

        #pragma unroll
        for (int cc = 0; cc < NCH; ++cc) {
            const float* xs = &s_x[cc][o0 * LSTR + tx];
            float a0 = 0.0f, a1 = 0.0f;
            #pragma unroll
            for (int i = 0; i < KS + 1; ++i) {          // 8 shared input rows
                float r[KS];
                #pragma unroll
                for (int j = 0; j < KS; ++j) r[j] = xs[i * LSTR + j];
                if (i < KS) {                           // taps for row A
                    #pragma unroll
                    for (int j = 0; j < KS; ++j)
                        a0 = fmaf(k0[i * KS + j], r[j], a0);
                }
                if (i >= 1) {                           // taps for row B
                    #pragma unroll
                    for (int j = 0; j < KS; ++j)
                        a1 = fmaf(k1[(i - 1) * KS + j], r[j], a1);
                }
            }
            const int c = cg * CGRP + cb * NCH + cc;
            out[(((size_t)b * CCH + c) * HH + hA) * WW + wcol] = a0;
            out[(((size_t)b * CCH + c) * HH + hB) * WW + wcol] = a1;
        }
        __syncthreads();
    }
}

// ---------------------------------------------------------------------------
extern "C" void kernel_launch(void* const* d_in, const int* in_sizes, int n_in,
                              void* d_out, int out_size, void* d_ws, size_t ws_size,
                              hipStream_t stream) {
    (void)in_sizes; (void)n_in; (void)out_size; (void)ws_size;
    const float* x      = (const float*)d_in[0];   // (8,128,128,128)
    const float* angle  = (const float*)d_in[1];   // (8,128,128)
    const float* w1     = (const float*)d_in[2];   // (8,2)
    const float* b1     = (const float*)d_in[3];   // (8,)
    const float* w2     = (const float*)d_in[4];   // (4,8)
    const float* b2     = (const float*)d_in[5];   // (4,)
    const float* bker   = (const float*)d_in[6];   // (4,7,7)
    float* outp = (float*)d_out;
    float* wgt  = (float*)d_ws;                    // B*H*W*4 floats = 2 MB

    // 131072 pixels, 16 pixels/wave, 8 waves/block -> 1024 blocks
    gate_weights_kernel<<<(BB * HH * WW) / 128, 256, 0, stream>>>(
        angle, w1, b1, w2, b2, wgt);

    // 8 W-tiles * 4 H-tiles * 8 batches * 4 channel-groups = 1024 blocks
    dirconv_kernel<<<1024, 256, 0, stream>>>(x, bker, wgt, outp);
}